// GraphPool_68650757259536
// MI455X (gfx1250) — compile-verified
//
#include <hip/hip_runtime.h>
#include <math.h>

#define BB   4
#define NN   4096
#define SS   1024
#define KK   32
#define DD   64
#define KS   10      // as_neighbor - 1 (as_neighbor = 11 per setup_inputs)
#define COUTC 128

typedef float v2f __attribute__((ext_vector_type(2)));
typedef float v8f __attribute__((ext_vector_type(8)));

// ---------------------------------------------------------------------------
// Generic fp32 WMMA GEMM:  C[M,N] = A[M,K] * W^T  (+ optional per-col bias)
// W element [n,k] at W[n*wsn + k*wsk]; WUNIT=true specializes wsk==1 so the
// B-operand pair merges into a single global_load_b64.
// One wave computes a 16x32 tile: one A fragment feeds two WMMA accumulators
// (alternating accs breaks the WMMA->WMMA RAW chain on the same register).
// Requires: M % 16 == 0, N % 32 == 0 (true for all call sites).
// Main K-loop is unguarded (EXEC all-ones); only the final partial K-step
// (Kd % 4 != 0, i.e. Kd in {3,69,134}) takes the guarded tail path.
// A 16x4 f32 layout: lane m = lane&15, K = (k0 + vgpr + 2*(lane>>4))
// B 4x16 f32 layout: lane n = lane&15, K = (k0 + vgpr + 2*(lane>>4))
// C/D 16x16 f32:     row = v + 8*(lane>>4), col = lane&15
// ---------------------------------------------------------------------------
template <bool WUNIT>
__global__ __launch_bounds__(32)
void gemm_f32_wmma(const float* __restrict__ A, const float* __restrict__ W,
                   const float* __restrict__ bias, float* __restrict__ C,
                   int M, int Kd, int Nd, int wsn, int wsk)
{
    int tiles_n = Nd >> 5;               // 32-wide column tiles
    int mt = blockIdx.x / tiles_n;
    int nt = blockIdx.x % tiles_n;
    int lane = threadIdx.x;
    int half = lane >> 4;
    int lm   = lane & 15;
    int row  = mt * 16 + lm;             // A row for this lane
    int col0 = nt * 32 + lm;             // B/C columns for this lane
    int col1 = col0 + 16;

    const float* Arow = A + (size_t)row * Kd;
    const float* Wc0  = W + (size_t)col0 * wsn;
    const float* Wc1  = W + (size_t)col1 * wsn;

    v8f acc0 = {0.f,0.f,0.f,0.f,0.f,0.f,0.f,0.f};
    v8f acc1 = {0.f,0.f,0.f,0.f,0.f,0.f,0.f,0.f};
    int kmain = Kd & ~3;
#pragma unroll 2
    for (int k0 = 0; k0 < kmain; k0 += 4) {
        int ka = k0 + 2 * half;
        v2f a, b0, b1;
        a.x = Arow[ka];
        a.y = Arow[ka + 1];
        if (WUNIT) {
            b0.x = Wc0[ka]; b0.y = Wc0[ka + 1];
            b1.x = Wc1[ka]; b1.y = Wc1[ka + 1];
        } else {
            b0.x = Wc0[(size_t)(ka    ) * wsk]; b0.y = Wc0[(size_t)(ka + 1) * wsk];
            b1.x = Wc1[(size_t)(ka    ) * wsk]; b1.y = Wc1[(size_t)(ka + 1) * wsk];
        }
        acc0 = __builtin_amdgcn_wmma_f32_16x16x4_f32(false, a, false, b0,
                                                     (short)0, acc0, false, false);
        acc1 = __builtin_amdgcn_wmma_f32_16x16x4_f32(false, a, false, b1,
                                                     (short)0, acc1, false, false);
    }
    if (kmain < Kd) {   // single guarded tail step (Kd in {3,69,134})
        int ka = kmain + 2 * half;
        v2f a, b0, b1;
        bool v0 = (ka < Kd), v1 = (ka + 1 < Kd);
        a.x  = v0 ? Arow[ka    ] : 0.f;
        a.y  = v1 ? Arow[ka + 1] : 0.f;
        b0.x = v0 ? Wc0[(size_t)(ka    ) * wsk] : 0.f;
        b0.y = v1 ? Wc0[(size_t)(ka + 1) * wsk] : 0.f;
        b1.x = v0 ? Wc1[(size_t)(ka    ) * wsk] : 0.f;
        b1.y = v1 ? Wc1[(size_t)(ka + 1) * wsk] : 0.f;
        acc0 = __builtin_amdgcn_wmma_f32_16x16x4_f32(false, a, false, b0,
                                                     (short)0, acc0, false, false);
        acc1 = __builtin_amdgcn_wmma_f32_16x16x4_f32(false, a, false, b1,
                                                     (short)0, acc1, false, false);
    }
    float bv0 = bias ? bias[col0] : 0.f;
    float bv1 = bias ? bias[col1] : 0.f;
#pragma unroll
    for (int v = 0; v < 8; ++v) {
        int r = mt * 16 + v + 8 * half;
        C[(size_t)r * Nd + col0] = acc0[v] + bv0;
        C[(size_t)r * Nd + col1] = acc1[v] + bv1;
    }
}

// ---------------------------------------------------------------------------
// BatchNorm pass 1: per-channel sum / sumsq (channel = index % C), C <= 128
// ---------------------------------------------------------------------------
__global__ __launch_bounds__(256)
void bn_stats(const float* __restrict__ x, float* __restrict__ stats,
              long long total, int C)
{
    __shared__ float ssum[128];
    __shared__ float ssq[128];
    for (int c = threadIdx.x; c < C; c += 256) { ssum[c] = 0.f; ssq[c] = 0.f; }
    __syncthreads();
    long long stride = (long long)gridDim.x * 256;
    for (long long i = (long long)blockIdx.x * 256 + threadIdx.x; i < total; i += stride) {
        float v = x[i];
        int c = (int)(i % C);
        atomicAdd(&ssum[c], v);
        atomicAdd(&ssq[c], v * v);
    }
    __syncthreads();
    for (int c = threadIdx.x; c < C; c += 256) {
        atomicAdd(&stats[c], ssum[c]);
        atomicAdd(&stats[C + c], ssq[c]);
    }
}

// BatchNorm pass 2 (in place): y = g*(x-m)*rsqrt(var+1e-5)+b ; act 0=none 1=relu 2=lrelu0.2
__global__ __launch_bounds__(256)
void bn_apply(float* __restrict__ x, const float* __restrict__ stats,
              const float* __restrict__ g, const float* __restrict__ be,
              long long total, int C, float invCount, int act)
{
    long long stride = (long long)gridDim.x * 256;
    for (long long i = (long long)blockIdx.x * 256 + threadIdx.x; i < total; i += stride) {
        int c = (int)(i % C);
        float m   = stats[c] * invCount;
        float var = stats[C + c] * invCount - m * m;
        float y = g[c] * (x[i] - m) * rsqrtf(var + 1e-5f) + be[c];
        if (act == 1) y = fmaxf(y, 0.f);
        else if (act == 2) y = (y > 0.f) ? y : 0.2f * y;
        x[i] = y;
    }
}

// ---------------------------------------------------------------------------
// Gather centers: newxyz[bs,3] and write output 0 as (B,3,S)
// ---------------------------------------------------------------------------
__global__ __launch_bounds__(256)
void k_gather_center(const float* __restrict__ xyz, const int* __restrict__ cidx,
                     float* __restrict__ newxyz, float* __restrict__ out0)
{
    int i = blockIdx.x * 256 + threadIdx.x;
    if (i >= BB * SS * 3) return;
    int c = i % 3, bs = i / 3, b = bs / SS, s = bs % SS;
    int id = cidx[bs];
    float v = xyz[(size_t)b * 3 * NN + (size_t)c * NN + id];
    newxyz[i] = v;
    out0[(size_t)b * 3 * SS + (size_t)c * SS + s] = v;
}

// ---------------------------------------------------------------------------
// Fused distance + stable top-k selection (ascending, ties -> lowest index,
// matching jnp.argsort). query==nullptr -> self-kNN (query = point `s`).
// One 256-thread block per query row; N=4096 dists live in LDS.
// ---------------------------------------------------------------------------
__global__ __launch_bounds__(256)
void knn_select(const float* __restrict__ xyz, const float* __restrict__ query,
                int* __restrict__ outIdx, int nsel, int skipFirst, int Srows)
{
    __shared__ float dist[NN];
    __shared__ float rmv[256];
    __shared__ int   rmi[256];
    int row = blockIdx.x;
    int b = row / Srows, s = row % Srows;
    int tid = threadIdx.x;
    const float* X = xyz + (size_t)b * 3 * NN;
    float q0, q1, q2;
    if (query) { q0 = query[row*3+0]; q1 = query[row*3+1]; q2 = query[row*3+2]; }
    else       { q0 = X[s]; q1 = X[NN + s]; q2 = X[2*NN + s]; }
    for (int n = tid; n < NN; n += 256) {
        float dx = q0 - X[n], dy = q1 - X[NN + n], dz = q2 - X[2*NN + n];
        dist[n] = dx*dx + dy*dy + dz*dz;
    }
    __syncthreads();
    for (int sel = 0; sel < nsel; ++sel) {
        float bm = 3.4e38f; int bi = NN;
        for (int n = tid; n < NN; n += 256) {
            float d = dist[n];
            if (d < bm || (d == bm && n < bi)) { bm = d; bi = n; }
        }
        rmv[tid] = bm; rmi[tid] = bi;
        __syncthreads();
        for (int off = 128; off > 0; off >>= 1) {
            if (tid < off) {
                float dv = rmv[tid + off]; int di = rmi[tid + off];
                if (dv < rmv[tid] || (dv == rmv[tid] && di < rmi[tid])) {
                    rmv[tid] = dv; rmi[tid] = di;
                }
            }
            __syncthreads();
        }
        if (tid == 0) {
            int w = rmi[0];
            if (sel >= skipFirst)
                outIdx[(size_t)row * (nsel - skipFirst) + sel - skipFirst] = w;
            dist[w] = 3.4e38f;
        }
        __syncthreads();
    }
}

// diff[b,n,j,c] = x[gidxs[b,n,j]][c] - x[n][c]
__global__ __launch_bounds__(256)
void k_diff(const float* __restrict__ xyz, const int* __restrict__ gidxs,
            float* __restrict__ diff)
{
    int r = blockIdx.x * 256 + threadIdx.x;
    if (r >= BB * NN * KS) return;
    int b = r / (NN * KS);
    int n = (r / KS) % NN;
    int g = gidxs[r];
    const float* X = xyz + (size_t)b * 3 * NN;
    diff[r*3+0] = X[g]        - X[n];
    diff[r*3+1] = X[NN + g]   - X[NN + n];
    diff[r*3+2] = X[2*NN + g] - X[2*NN + n];
}

// bvout[r,o] = sum_c bvconv_w[o,c]*diff[r,c]   (3x3)
__global__ __launch_bounds__(256)
void k_bvconv(const float* __restrict__ diff, const float* __restrict__ w,
              float* __restrict__ bvout)
{
    int r = blockIdx.x * 256 + threadIdx.x;
    if (r >= BB * NN * KS) return;
    float d0 = diff[r*3+0], d1 = diff[r*3+1], d2 = diff[r*3+2];
#pragma unroll
    for (int o = 0; o < 3; ++o)
        bvout[r*3+o] = w[o*3+0]*d0 + w[o*3+1]*d1 + w[o*3+2]*d2;
}

// basev[p,c] = mean_j bvout[(p*KS+j),c]
__global__ __launch_bounds__(256)
void k_basev(const float* __restrict__ bvout, float* __restrict__ basev)
{
    int i = blockIdx.x * 256 + threadIdx.x;
    if (i >= BB * NN * 3) return;
    int c = i % 3, p = i / 3;
    float s = 0.f;
    for (int j = 0; j < KS; ++j) s += bvout[((size_t)p * KS + j) * 3 + c];
    basev[i] = s * (1.f / KS);
}

// d1f[r] = sum_c bconv[c]*diff[r,c]
__global__ __launch_bounds__(256)
void k_d1f(const float* __restrict__ diff, const float* __restrict__ w,
           float* __restrict__ d1f)
{
    int r = blockIdx.x * 256 + threadIdx.x;
    if (r >= BB * NN * KS) return;
    d1f[r] = w[0]*diff[r*3+0] + w[1]*diff[r*3+1] + w[2]*diff[r*3+2];
}

// t[r] = bww . [|d0|,|d1|,|d2|, d1f[r]]
__global__ __launch_bounds__(256)
void k_tconv(const float* __restrict__ diff, const float* __restrict__ d1f,
             const float* __restrict__ w, float* __restrict__ t)
{
    int r = blockIdx.x * 256 + threadIdx.x;
    if (r >= BB * NN * KS) return;
    t[r] = w[0]*fabsf(diff[r*3+0]) + w[1]*fabsf(diff[r*3+1])
         + w[2]*fabsf(diff[r*3+2]) + w[3]*d1f[r];
}

// cos[r] = (basev . diff) / sqrt(|basev|^2 * |diff|^2 + 1e-10)
__global__ __launch_bounds__(256)
void k_cos(const float* __restrict__ basev, const float* __restrict__ diff,
           float* __restrict__ cosb)
{
    int r = blockIdx.x * 256 + threadIdx.x;
    if (r >= BB * NN * KS) return;
    int p = r / KS;
    float b0 = basev[p*3+0], b1 = basev[p*3+1], b2 = basev[p*3+2];
    float d0 = diff[r*3+0],  d1 = diff[r*3+1],  d2 = diff[r*3+2];
    float dd1 = b0*b0 + b1*b1 + b2*b2;
    float dd2 = d0*d0 + d1*d1 + d2*d2;
    float inn = b0*d0 + b1*d1 + b2*d2;
    cosb[r] = inn / sqrtf(dd1 * dd2 + 1e-10f);
}

// aggsum[p,c5] = sum_j cos * [ |diff|(3), d1f, t ]
__global__ __launch_bounds__(256)
void k_aggsum(const float* __restrict__ diff, const float* __restrict__ d1f,
              const float* __restrict__ t, const float* __restrict__ cosb,
              float* __restrict__ aggsum)
{
    int i = blockIdx.x * 256 + threadIdx.x;
    if (i >= BB * NN * 5) return;
    int c = i % 5, p = i / 5;
    float s = 0.f;
    for (int j = 0; j < KS; ++j) {
        int r = p * KS + j;
        float v = (c < 3) ? fabsf(diff[r*3+c]) : ((c == 3) ? d1f[r] : t[r]);
        s += cosb[r] * v;
    }
    aggsum[i] = s;
}

// agg69[p,c] = c<64 ? points[b,c,n] : aggsum[p, c-64]
__global__ __launch_bounds__(256)
void k_agg69(const float* __restrict__ points, const float* __restrict__ aggsum,
             float* __restrict__ agg69)
{
    int i = blockIdx.x * 256 + threadIdx.x;
    if (i >= BB * NN * 69) return;
    int c = i % 69, p = i / 69, b = p / NN, n = p % NN;
    agg69[i] = (c < DD) ? points[(size_t)b * DD * NN + (size_t)c * NN + n]
                        : aggsum[(size_t)p * 5 + (c - DD)];
}

// skipmax[bs,c134]: c<67 -> max_j(ngf-ncf); c>=67 -> ncf
__global__ __launch_bounds__(256)
void k_skipmax(const float* __restrict__ xyz, const float* __restrict__ points,
               const int* __restrict__ cidx, const int* __restrict__ gidx,
               float* __restrict__ sm)
{
    int i = blockIdx.x * 256 + threadIdx.x;
    if (i >= BB * SS * 134) return;
    int c = i % 134, bs = i / 134, b = bs / SS;
    int ci = cidx[bs];
    int cc = (c < 67) ? c : (c - 67);
    float ncf = (cc < 3) ? xyz[(size_t)b*3*NN + (size_t)cc*NN + ci]
                         : points[(size_t)b*DD*NN + (size_t)(cc-3)*NN + ci];
    float outv;
    if (c >= 67) outv = ncf;
    else {
        float m = -3.4e38f;
        for (int j = 0; j < KK; ++j) {
            int g = gidx[bs * KK + j];
            float ng = (cc < 3) ? xyz[(size_t)b*3*NN + (size_t)cc*NN + g]
                                : points[(size_t)b*DD*NN + (size_t)(cc-3)*NN + g];
            m = fmaxf(m, ng - ncf);
        }
        outv = m;
    }
    sm[i] = outv;
}

// dst[(b*nPer+t), c64] = src[(b*NN + idx[b*nPer+t]), c64]
__global__ __launch_bounds__(256)
void k_gather_rows(const float* __restrict__ src, const int* __restrict__ idx,
                   int nPer, float* __restrict__ dst, long long total)
{
    long long i = (long long)blockIdx.x * 256 + threadIdx.x;
    if (i >= total) return;
    int c = (int)(i % DD);
    long long t = i / DD;
    int b = (int)(t / nPer);
    dst[i] = src[((size_t)b * NN + idx[t]) * DD + c];
}

// posin[r,3] = gxyz - newxyz
__global__ __launch_bounds__(256)
void k_posin(const float* __restrict__ xyz, const int* __restrict__ gidx,
             const float* __restrict__ newxyz, float* __restrict__ posin)
{
    int i = blockIdx.x * 256 + threadIdx.x;
    if (i >= BB * SS * KK * 3) return;
    int c = i % 3, r = i / 3, bs = r / KK, b = bs / SS;
    int g = gidx[r];
    posin[i] = xyz[(size_t)b*3*NN + (size_t)c*NN + g] - newxyz[bs*3 + c];
}

// qk1[r] = dot64(w2[bs], gfeat[r])
__global__ __launch_bounds__(256)
void k_qk1(const float* __restrict__ w2, const float* __restrict__ gfeat,
           float* __restrict__ qk1)
{
    int r = blockIdx.x * 256 + threadIdx.x;
    if (r >= BB * SS * KK) return;
    int bs = r / KK;
    const float* wv = w2 + (size_t)bs * DD;
    const float* gv = gfeat + (size_t)r * DD;
    float s = 0.f;
    for (int c = 0; c < DD; ++c) s += wv[c] * gv[c];
    qk1[r] = s;
}

// In place: KL[i] = Ql - KL + qk1*ch_w + pos   (attention input)
__global__ __launch_bounds__(256)
void k_combine(float* __restrict__ KL, const float* __restrict__ Ql,
               const float* __restrict__ qk1, const float* __restrict__ chw,
               const float* __restrict__ posb, long long total)
{
    long long i = (long long)blockIdx.x * 256 + threadIdx.x;
    if (i >= total) return;
    int c = (int)(i % COUTC);
    long long r = i / COUTC;
    int bs = (int)(r / KK);
    KL[i] = Ql[(size_t)bs * COUTC + c] - KL[i] + qk1[r] * chw[c] + posb[i];
}

// logits + softmax over K=32 (one wave per (b,s); lane = k)
__global__ __launch_bounds__(32)
void k_softmax(const float* __restrict__ attOut, const float* __restrict__ w1,
               const float* __restrict__ b1, float* __restrict__ attw)
{
    int bs = blockIdx.x;
    int k = threadIdx.x;
    const float* row = attOut + ((size_t)bs * KK + k) * COUTC;
    float lg = 0.f;
    for (int c = 0; c < COUTC; ++c) lg += w1[c] * row[c];
    lg = (lg + b1[0]) * 0.08838834764831845f;  // / sqrt(128)
    float m = lg;
    for (int off = 16; off > 0; off >>= 1) m = fmaxf(m, __shfl_xor(m, off, 32));
    float e = __expf(lg - m);
    float s = e;
    for (int off = 16; off > 0; off >>= 1) s += __shfl_xor(s, off, 32);
    attw[(size_t)bs * KK + k] = e / s;
}

// out0[bs,c] = sum_k att * (Vl + pos)
__global__ __launch_bounds__(256)
void k_outsum(const float* __restrict__ attw, const float* __restrict__ VL,
              const float* __restrict__ posb, float* __restrict__ out0)
{
    int i = blockIdx.x * 256 + threadIdx.x;
    if (i >= BB * SS * COUTC) return;
    int c = i % COUTC, bs = i / COUTC;
    float s = 0.f;
    for (int j = 0; j < KK; ++j) {
        size_t r = (size_t)bs * KK + j;
        s += attw[r] * (VL[r * COUTC + c] + posb[r * COUTC + c]);
    }
    out0[i] = s;
}

// ffin = [pcout(128) | skip(128)]
__global__ __launch_bounds__(256)
void k_ffin(const float* __restrict__ pcout, const float* __restrict__ skipb,
            float* __restrict__ ffin)
{
    int i = blockIdx.x * 256 + threadIdx.x;
    if (i >= BB * SS * 256) return;
    int c = i % 256, bs = i / 256;
    ffin[i] = (c < COUTC) ? pcout[(size_t)bs * COUTC + c]
                          : skipb[(size_t)bs * COUTC + (c - COUTC)];
}

// out1 (B,128,S) <- ffout [bs,128]
__global__ __launch_bounds__(256)
void k_store(const float* __restrict__ ffout, float* __restrict__ out1)
{
    int i = blockIdx.x * 256 + threadIdx.x;
    if (i >= BB * SS * COUTC) return;
    int c = i % COUTC, bs = i / COUTC, b = bs / SS, s = bs % SS;
    out1[((size_t)b * COUTC + c) * SS + s] = ffout[i];
}

// ---------------------------------------------------------------------------
extern "C" void kernel_launch(void* const* d_in, const int* in_sizes, int n_in,
                              void* d_out, int out_size, void* d_ws, size_t ws_size,
                              hipStream_t stream)
{
    (void)in_sizes; (void)n_in; (void)out_size;
    const float* xyz      = (const float*)d_in[0];
    const float* points   = (const float*)d_in[1];
    const int*   cidx     = (const int*)  d_in[2];
    const float* bvconv_w = (const float*)d_in[4];
    const float* bvbn_g   = (const float*)d_in[5];
    const float* bvbn_b   = (const float*)d_in[6];
    const float* bconv_w  = (const float*)d_in[7];
    const float* bbn_g    = (const float*)d_in[8];
    const float* bbn_b    = (const float*)d_in[9];
    const float* bww_w    = (const float*)d_in[10];
    const float* bwbn_g   = (const float*)d_in[11];
    const float* bwbn_b   = (const float*)d_in[12];
    const float* wga01_w  = (const float*)d_in[13];
    const float* wga_g    = (const float*)d_in[14];
    const float* wga_b    = (const float*)d_in[15];
    const float* lq_w     = (const float*)d_in[16];
    const float* lkv_w    = (const float*)d_in[17];
    const float* l1q_w    = (const float*)d_in[18];
    const float* l1k_w    = (const float*)d_in[19];
    const float* ch_w     = (const float*)d_in[20];
    const float* atw_w    = (const float*)d_in[21];
    const float* atw_b    = (const float*)d_in[22];
    const float* atw_g    = (const float*)d_in[23];
    const float* atw_be   = (const float*)d_in[24];
    const float* atw1_w   = (const float*)d_in[25];
    const float* atw1_b   = (const float*)d_in[26];
    const float* pc_w     = (const float*)d_in[27];
    const float* pc_b     = (const float*)d_in[28];
    const float* pc_g     = (const float*)d_in[29];
    const float* pc_be    = (const float*)d_in[30];
    const float* sk_w     = (const float*)d_in[31];
    const float* sk_g     = (const float*)d_in[32];
    const float* sk_b     = (const float*)d_in[33];
    const float* ff_w     = (const float*)d_in[34];
    const float* ff_g     = (const float*)d_in[35];
    const float* ff_b     = (const float*)d_in[36];
    const float* l_w0     = (const float*)d_in[37];
    const float* l_b0     = (const float*)d_in[38];
    const float* l_g0     = (const float*)d_in[39];
    const float* l_be0    = (const float*)d_in[40];
    const float* l_w1     = (const float*)d_in[41];
    const float* l_b1     = (const float*)d_in[42];
    const float* l_g1     = (const float*)d_in[43];
    const float* l_be1    = (const float*)d_in[44];
    const float* l_w2     = (const float*)d_in[45];
    const float* l_b2     = (const float*)d_in[46];
    const float* l_g2     = (const float*)d_in[47];
    const float* l_be2    = (const float*)d_in[48];

    float* out0 = (float*)d_out;                 // (B,3,S)
    float* out1 = (float*)d_out + BB * 3 * SS;   // (B,128,S)

    // ---- workspace bump allocator ----
    char* base = (char*)d_ws;
    size_t cur = 0;
    auto alloc = [&](size_t bytes) -> void* {
        void* p = base + cur;
        cur = (cur + bytes + 255) & ~(size_t)255;
        return p;
    };
    float* newxyz = (float*)alloc((size_t)BB*SS*3*4);
    int*   gidx   = (int*)  alloc((size_t)BB*SS*KK*4);
    int*   gidxs  = (int*)  alloc((size_t)BB*NN*KS*4);
    float* diff   = (float*)alloc((size_t)BB*NN*KS*3*4);
    float* bvout  = (float*)alloc((size_t)BB*NN*KS*3*4);
    float* basev  = (float*)alloc((size_t)BB*NN*3*4);
    float* d1f    = (float*)alloc((size_t)BB*NN*KS*4);
    float* tbuf   = (float*)alloc((size_t)BB*NN*KS*4);
    float* cosb   = (float*)alloc((size_t)BB*NN*KS*4);
    float* aggsum = (float*)alloc((size_t)BB*NN*5*4);
    float* agg69  = (float*)alloc((size_t)BB*NN*69*4);
    float* aggF   = (float*)alloc((size_t)BB*NN*DD*4);
    float* skipmx = (float*)alloc((size_t)BB*SS*134*4);
    float* skipb  = (float*)alloc((size_t)BB*SS*COUTC*4);
    float* aggc   = (float*)alloc((size_t)BB*SS*DD*4);
    float* Ql     = (float*)alloc((size_t)BB*SS*COUTC*4);
    float* Ql1    = (float*)alloc((size_t)BB*SS*COUTC*4);
    float* w2b    = (float*)alloc((size_t)BB*SS*DD*4);
    float* gfeat  = (float*)alloc((size_t)BB*SS*KK*DD*4);
    float* KL     = (float*)alloc((size_t)BB*SS*KK*COUTC*4);
    float* VL     = (float*)alloc((size_t)BB*SS*KK*COUTC*4);
    float* qk1    = (float*)alloc((size_t)BB*SS*KK*4);
    float* posin  = (float*)alloc((size_t)BB*SS*KK*3*4);
    float* ph1    = (float*)alloc((size_t)BB*SS*KK*DD*4);
    float* ph2    = (float*)alloc((size_t)BB*SS*KK*DD*4);
    float* posb   = (float*)alloc((size_t)BB*SS*KK*COUTC*4);
    float* attOut = (float*)alloc((size_t)BB*SS*KK*COUTC*4);
    float* attw   = (float*)alloc((size_t)BB*SS*KK*4);
    float* outs   = (float*)alloc((size_t)BB*SS*COUTC*4);
    float* pcout  = (float*)alloc((size_t)BB*SS*COUTC*4);
    float* ffin   = (float*)alloc((size_t)BB*SS*256*4);
    float* ffout  = (float*)alloc((size_t)BB*SS*COUTC*4);
    float* stats  = (float*)alloc((size_t)10*256*4);   // 10 BN slots x (sum|sumsq)
    if (cur > ws_size) return;  // insufficient scratch (compile-only environment)

    hipMemsetAsync(stats, 0, (size_t)10*256*4, stream);

    auto blocks = [](long long n) { return dim3((unsigned)((n + 255) / 256)); };
    auto gemm = [&](const float* A, const float* W, const float* bias, float* C,
                    int M, int Kd, int Nd, int wsn, int wsk) {
        dim3 g((unsigned)((M / 16) * (Nd / 32)));
        if (wsk == 1)
            gemm_f32_wmma<true ><<<g, 32, 0, stream>>>(A, W, bias, C, M, Kd, Nd, wsn, wsk);
        else
            gemm_f32_wmma<false><<<g, 32, 0, stream>>>(A, W, bias, C, M, Kd, Nd, wsn, wsk);
    };
    auto bn = [&](float* buf, long long total, int C, const float* g,
                  const float* be, float invCnt, int act, int slot) {
        unsigned gs = (unsigned)(((total + 255) / 256) < 8192 ? ((total + 255) / 256) : 8192);
        bn_stats<<<dim3(gs), 256, 0, stream>>>(buf, stats + slot*256, total, C);
        bn_apply<<<dim3(gs), 256, 0, stream>>>(buf, stats + slot*256, g, be,
                                               total, C, invCnt, act);
    };

    const long long Tnk = (long long)BB*NN*KS;      // 163840
    const long long Tsk = (long long)BB*SS*KK;      // 131072

    // ---- grouping ----
    k_gather_center<<<blocks(BB*SS*3), 256, 0, stream>>>(xyz, cidx, newxyz, out0);
    knn_select<<<dim3(BB*SS), 256, 0, stream>>>(xyz, newxyz, gidx, KK, 0, SS);
    knn_select<<<dim3(BB*NN), 256, 0, stream>>>(xyz, nullptr, gidxs, KS + 1, 1, NN);
    k_diff<<<blocks(Tnk), 256, 0, stream>>>(xyz, gidxs, diff);

    // ---- base-vector / cosine path ----
    k_bvconv<<<blocks(Tnk), 256, 0, stream>>>(diff, bvconv_w, bvout);
    bn(bvout, Tnk*3, 3, bvbn_g, bvbn_b, 1.f/(float)Tnk, 2, 0);
    k_basev<<<blocks((long long)BB*NN*3), 256, 0, stream>>>(bvout, basev);
    k_d1f<<<blocks(Tnk), 256, 0, stream>>>(diff, bconv_w, d1f);
    bn(d1f, Tnk, 1, bbn_g, bbn_b, 1.f/(float)Tnk, 2, 1);
    k_tconv<<<blocks(Tnk), 256, 0, stream>>>(diff, d1f, bww_w, tbuf);
    bn(tbuf, Tnk, 1, bwbn_g, bwbn_b, 1.f/(float)Tnk, 2, 2);
    k_cos<<<blocks(Tnk), 256, 0, stream>>>(basev, diff, cosb);
    k_aggsum<<<blocks((long long)BB*NN*5), 256, 0, stream>>>(diff, d1f, tbuf, cosb, aggsum);

    // ---- per-point feature GEMM (wga: 69 -> 64) ----
    k_agg69<<<blocks((long long)BB*NN*69), 256, 0, stream>>>(points, aggsum, agg69);
    gemm(agg69, wga01_w, nullptr, aggF, BB*NN, 69, DD, 69, 1);
    bn(aggF, (long long)BB*NN*DD, DD, wga_g, wga_b, 1.f/(float)(BB*NN), 1, 3);

    // ---- skip path (max-pool + sk: 134 -> 128) ----
    k_skipmax<<<blocks((long long)BB*SS*134), 256, 0, stream>>>(xyz, points, cidx, gidx, skipmx);
    gemm(skipmx, sk_w, nullptr, skipb, BB*SS, 134, COUTC, 134, 1);
    bn(skipb, (long long)BB*SS*COUTC, COUTC, sk_g, sk_b, 1.f/(float)(BB*SS), 1, 4);

    // ---- gathers + Q/K/V GEMMs ----
    k_gather_rows<<<blocks((long long)BB*SS*DD), 256, 0, stream>>>(aggF, cidx, SS, aggc, (long long)BB*SS*DD);
    k_gather_rows<<<blocks(Tsk*DD), 256, 0, stream>>>(aggF, gidx, SS*KK, gfeat, Tsk*DD);
    gemm(aggc, lq_w,  nullptr, Ql,  BB*SS, DD, COUTC, DD, 1);
    gemm(aggc, l1q_w, nullptr, Ql1, BB*SS, DD, COUTC, DD, 1);
    // w2 = l1k_w^T * Ql1  (avoids materializing Kl1: qk1 = w2 . gfeat)
    gemm(Ql1, l1k_w, nullptr, w2b, BB*SS, COUTC, DD, 1, DD);
    gemm(gfeat, lkv_w,              nullptr, KL, (int)Tsk, DD, COUTC, DD, 1);
    gemm(gfeat, lkv_w + COUTC*DD,   nullptr, VL, (int)Tsk, DD, COUTC, DD, 1);
    k_qk1<<<blocks(Tsk), 256, 0, stream>>>(w2b, gfeat, qk1);

    // ---- positional MLP 3 -> 64 -> 64 -> 128 ----
    k_posin<<<blocks(Tsk*3), 256, 0, stream>>>(xyz, gidx, newxyz, posin);
    gemm(posin, l_w0, l_b0, ph1, (int)Tsk, 3, DD, 3, 1);
    bn(ph1, Tsk*DD, DD, l_g0, l_be0, 1.f/(float)Tsk, 1, 5);
    gemm(ph1, l_w1, l_b1, ph2, (int)Tsk, DD, DD, DD, 1);
    bn(ph2, Tsk*DD, DD, l_g1, l_be1, 1.f/(float)Tsk, 1, 6);
    gemm(ph2, l_w2, l_b2, posb, (int)Tsk, DD, COUTC, DD, 1);
    (void)l_g2; (void)l_be2;  // last MLP layer has no BN in reference

    // ---- attention ----
    k_combine<<<blocks(Tsk*COUTC), 256, 0, stream>>>(KL, Ql, qk1, ch_w, posb, Tsk*COUTC);
    gemm(KL, atw_w, atw_b, attOut, (int)Tsk, COUTC, COUTC, COUTC, 1);
    bn(attOut, Tsk*COUTC, COUTC, atw_g, atw_be, 1.f/(float)Tsk, 1, 7);
    k_softmax<<<dim3(BB*SS), 32, 0, stream>>>(attOut, atw1_w, atw1_b, attw);
    k_outsum<<<blocks((long long)BB*SS*COUTC), 256, 0, stream>>>(attw, VL, posb, outs);

    // ---- post conv + feed-forward ----
    gemm(outs, pc_w, pc_b, pcout, BB*SS, COUTC, COUTC, COUTC, 1);
    bn(pcout, (long long)BB*SS*COUTC, COUTC, pc_g, pc_be, 1.f/(float)(BB*SS), 1, 8);
    k_ffin<<<blocks((long long)BB*SS*256), 256, 0, stream>>>(pcout, skipb, ffin);
    gemm(ffin, ff_w, nullptr, ffout, BB*SS, 256, COUTC, 256, 1);
    bn(ffout, (long long)BB*SS*COUTC, COUTC, ff_g, ff_b, 1.f/(float)(BB*SS), 1, 9);
    k_store<<<blocks((long long)BB*SS*COUTC), 256, 0, stream>>>(ffout, out1);
}